// SIGReg_12799002542242
// MI455X (gfx1250) — compile-verified
//
#include <hip/hip_runtime.h>

typedef __attribute__((ext_vector_type(2))) float v2f;
typedef __attribute__((ext_vector_type(8))) float v8f;

#define T_DIM 32
#define B_DIM 64
#define D_DIM 256
#define P_DIM 1024
#define KNOTS 17
#define DT (3.0f / 16.0f)   // knot spacing: 3/(KNOTS-1)

// ---------------------------------------------------------------------------
// Kernel 1: inverse column norms of A (D x P).  Thread per column p; loads of
// A[d*P + p] are coalesced across threads.  inv[p] = 1/max(||A[:,p]||, 1e-12).
// ---------------------------------------------------------------------------
__global__ void __launch_bounds__(256)
colnorm_kernel(const float* __restrict__ A, float* __restrict__ inv) {
    int p = blockIdx.x * blockDim.x + threadIdx.x;
    if (p >= P_DIM) return;
    float s = 0.0f;
    #pragma unroll 8
    for (int d = 0; d < D_DIM; ++d) {
        float a = A[d * P_DIM + p];
        s = fmaf(a, a, s);
    }
    float n = fmaxf(sqrtf(s), 1e-12f);
    inv[p] = 1.0f / n;
}

// ---------------------------------------------------------------------------
// Kernel 2: fused WMMA GEMM + characteristic-function statistic.
//
// Grid: 256 blocks x 256 threads = 2048 wave32 waves.  Wave w handles
//   t      = w / 64            (time slice: rows t*64 .. t*64+63 of proj@A)
//   ntile  = w % 64            (16 consecutive p columns)
// and computes a 64x16 tile of X = proj @ Anorm via 4 stacked 16x16 WMMA
// accumulators over K=256 (64 steps of V_WMMA_F32_16X16X4_F32), then reduces
// the CF statistic over the B dimension entirely in registers.
// ---------------------------------------------------------------------------
__global__ void __launch_bounds__(256)
gemm_cf_kernel(const float* __restrict__ proj, const float* __restrict__ A,
               const float* __restrict__ inv, float* __restrict__ partials) {
    __shared__ float wsum[8];

    const int lane = threadIdx.x & 31;
    const int wave = threadIdx.x >> 5;
    const int wgid = blockIdx.x * 8 + wave;   // 0..2047
    const int t    = wgid >> 6;               // 0..31
    const int ntl  = wgid & 63;               // 0..63
    const int half = lane >> 4;               // 0 or 1
    const int l16  = lane & 15;
    const int col  = ntl * 16 + l16;          // p column owned by this lane

    const float invc = inv[col];              // fold column normalization into B

    const float* projt = proj + (size_t)t * B_DIM * D_DIM;

    v8f acc[4];
    #pragma unroll
    for (int mt = 0; mt < 4; ++mt)
        #pragma unroll
        for (int r = 0; r < 8; ++r) acc[mt][r] = 0.0f;

    // K loop: 4 per WMMA.  A frag: lane l16 = row M, VGPRs hold K = kk, kk+1
    // with kk = k0 + 2*half  (8-byte aligned -> single b64 load).
    // B frag mirrors: VGPRs hold rows K = kk, kk+1 at column `col`.
    for (int k0 = 0; k0 < D_DIM; k0 += 4) {
        const int kk = k0 + 2 * half;

        v2f bf;
        bf[0] = A[(size_t)kk * P_DIM + col] * invc;
        bf[1] = A[(size_t)(kk + 1) * P_DIM + col] * invc;

        #pragma unroll
        for (int mt = 0; mt < 4; ++mt) {
            const float* ap = projt + (size_t)(mt * 16 + l16) * D_DIM + kk;
            v2f af = *(const v2f*)ap;
            acc[mt] = __builtin_amdgcn_wmma_f32_16x16x4_f32(
                /*neg_a=*/false, af, /*neg_b=*/false, bf,
                /*c_mod=*/(short)0, acc[mt],
                /*reuse_a=*/false, /*reuse_b=*/false);
        }
    }

    // --- CF statistic over this lane's 32 of the 64 B-values for column p ---
    // Chebyshev recurrence: one __cosf + one __sinf per x, then
    //   c_{k+1} = 2 c1 c_k - c_{k-1},  s_{k+1} = 2 c1 s_k - s_{k-1}
    float cs[KNOTS], ss[KNOTS];
    #pragma unroll
    for (int k = 0; k < KNOTS; ++k) { cs[k] = 0.0f; ss[k] = 0.0f; }

    #pragma unroll
    for (int mt = 0; mt < 4; ++mt) {
        #pragma unroll
        for (int r = 0; r < 8; ++r) {
            float x  = acc[mt][r];
            float th = x * DT;
            float c1 = __cosf(th);
            float s1 = __sinf(th);
            cs[0] += 1.0f;                 // cos(0)=1, sin(0)=0
            cs[1] += c1;  ss[1] += s1;
            float tc = 2.0f * c1;
            float cp = 1.0f, sp = 0.0f, ck = c1, sk = s1;
            #pragma unroll
            for (int k = 2; k < KNOTS; ++k) {
                float cn = fmaf(tc, ck, -cp);
                float sn = fmaf(tc, sk, -sp);
                cs[k] += cn; ss[k] += sn;
                cp = ck; sp = sk; ck = cn; sk = sn;
            }
        }
    }

    // Lanes L and L+16 together cover all 64 B-values of column p: merge.
    #pragma unroll
    for (int k = 0; k < KNOTS; ++k) {
        cs[k] += __shfl_xor(cs[k], 16, 32);
        ss[k] += __shfl_xor(ss[k], 16, 32);
    }

    // Weighted squared error vs phi(t_k); knot constants generated inline.
    const float invB = 1.0f / (float)B_DIM;
    float val = 0.0f;
    #pragma unroll
    for (int k = 0; k < KNOTS; ++k) {
        float tk   = (float)k * DT;
        float phik = __expf(-0.5f * tk * tk);
        float wk   = (k == 0 || k == KNOTS - 1) ? DT : 2.0f * DT;
        float ce   = cs[k] * invB - phik;
        float se   = ss[k] * invB;
        val += (wk * phik) * (ce * ce + se * se);
    }

    // Wave reduction; every p is duplicated in two lanes -> halve.
    #pragma unroll
    for (int off = 1; off < 32; off <<= 1) val += __shfl_xor(val, off, 32);
    val *= 0.5f;

    if (lane == 0) wsum[wave] = val;
    __syncthreads();
    if (threadIdx.x == 0) {
        float s = 0.0f;
        #pragma unroll
        for (int w = 0; w < 8; ++w) s += wsum[w];
        partials[blockIdx.x] = s;   // deterministic per-block partial
    }
}

// ---------------------------------------------------------------------------
// Kernel 3: deterministic final reduction of the 256 block partials.
// statistic = (sum over t,p of weighted err) * B / (T*P)
// ---------------------------------------------------------------------------
__global__ void __launch_bounds__(256)
reduce_kernel(const float* __restrict__ partials, float* __restrict__ out) {
    __shared__ float buf[256];
    buf[threadIdx.x] = partials[threadIdx.x];
    __syncthreads();
    #pragma unroll
    for (int s = 128; s > 0; s >>= 1) {
        if (threadIdx.x < s) buf[threadIdx.x] += buf[threadIdx.x + s];
        __syncthreads();
    }
    if (threadIdx.x == 0)
        out[0] = buf[0] * ((float)B_DIM / (float)(T_DIM * P_DIM));
}

// ---------------------------------------------------------------------------
extern "C" void kernel_launch(void* const* d_in, const int* in_sizes, int n_in,
                              void* d_out, int out_size, void* d_ws, size_t ws_size,
                              hipStream_t stream) {
    const float* proj = (const float*)d_in[0];   // (32, 64, 256) f32
    const float* A    = (const float*)d_in[1];   // (256, 1024) f32
    float* out = (float*)d_out;                  // scalar f32

    float* ws       = (float*)d_ws;
    float* inv      = ws;            // 1024 floats
    float* partials = ws + P_DIM;    // 256 floats   (total ~5 KB of d_ws)

    colnorm_kernel<<<P_DIM / 256, 256, 0, stream>>>(A, inv);
    gemm_cf_kernel<<<256, 256, 0, stream>>>(proj, A, inv, partials);
    reduce_kernel<<<1, 256, 0, stream>>>(partials, out);
}